// GraphKLLoss_79731772883225
// MI455X (gfx1250) — compile-verified
//
#include <hip/hip_runtime.h>
#include <math.h>

#define V_N   8192
#define D_FEAT 128
#define E_N   262144
#define K_N   512
#define DMAXC 0.5f
#define EPSW  1e-8f
#define EPSL  1e-5f

typedef float v2f __attribute__((ext_vector_type(2)));
typedef float v8f __attribute__((ext_vector_type(8)));

// ---------------------------------------------------------------- init ws
__global__ void k_init(float* __restrict__ Y, float* __restrict__ deg,
                       float* __restrict__ scal) {
    const size_t YN = (size_t)V_N * K_N;
    size_t stride = (size_t)gridDim.x * blockDim.x;
    for (size_t t = (size_t)blockIdx.x * blockDim.x + threadIdx.x;
         t < YN + V_N + 16; t += stride) {
        if (t < YN)            Y[t] = 0.0f;
        else if (t < YN + V_N) deg[t - YN] = 1.0f;   // identity contribution
        else                   scal[t - YN - V_N] = 0.0f;
    }
}

// -------------------------------------------------- PT = pmat^T  (V x K)
__global__ void k_transpose(const float* __restrict__ pmat, float* __restrict__ PT) {
    __shared__ float tile[32][33];
    int v0 = blockIdx.x * 32;
    int k0 = blockIdx.y * 32;
    tile[threadIdx.y][threadIdx.x] =
        pmat[(size_t)(k0 + threadIdx.y) * V_N + (v0 + threadIdx.x)];
    __syncthreads();
    PT[(size_t)(v0 + threadIdx.y) * K_N + (k0 + threadIdx.x)] =
        tile[threadIdx.x][threadIdx.y];
}

// ---------------- wave-per-edge: distance, weight, deg, scatter Y = S*PT
__global__ void k_edges(const float* __restrict__ x, const int* __restrict__ ei,
                        const float* __restrict__ PT, float* __restrict__ Y,
                        float* __restrict__ deg) {
    int wid  = (int)(((size_t)blockIdx.x * blockDim.x + threadIdx.x) >> 5);
    int lane = threadIdx.x & 31;
    if (wid >= E_N) return;
    int s = ei[wid];
    int t = ei[E_N + wid];

    const float4* xs = (const float4*)(x + (size_t)s * D_FEAT);
    const float4* xt = (const float4*)(x + (size_t)t * D_FEAT);
    float4 a = xs[lane];
    float4 b = xt[lane];
    float dx = a.x - b.x, dy = a.y - b.y, dz = a.z - b.z, dw = a.w - b.w;
    float pd = dx * dx + dy * dy + dz * dz + dw * dw;
#pragma unroll
    for (int off = 16; off; off >>= 1) pd += __shfl_xor(pd, off, 32);

    float w = (pd <= DMAXC) ? __expf(-pd) : EPSW;
    if (lane == 0) {
        atomicAdd(&deg[s], w);
        atomicAdd(&deg[t], w);
    }

    const float4* ps = (const float4*)(PT + (size_t)s * K_N);
    const float4* pt = (const float4*)(PT + (size_t)t * K_N);
    float* ys = Y + (size_t)s * K_N;
    float* yt = Y + (size_t)t * K_N;
    for (int k4 = lane; k4 < K_N / 4; k4 += 32) {
        float4 qs = ps[k4];
        float4 qt = pt[k4];
        int k = k4 * 4;
        atomicAdd(ys + k + 0, w * qt.x); atomicAdd(ys + k + 1, w * qt.y);
        atomicAdd(ys + k + 2, w * qt.z); atomicAdd(ys + k + 3, w * qt.w);
        atomicAdd(yt + k + 0, w * qs.x); atomicAdd(yt + k + 1, w * qs.y);
        atomicAdd(yt + k + 2, w * qs.z); atomicAdd(yt + k + 3, w * qs.w);
    }
}

// ---------------------------- Z = diag(deg-1)*PT - Y   (in place into Y)
__global__ void k_z(float* __restrict__ Y, const float* __restrict__ PT,
                    const float* __restrict__ deg) {
    const size_t N = (size_t)V_N * K_N;
    size_t stride = (size_t)gridDim.x * blockDim.x;
    for (size_t i = (size_t)blockIdx.x * blockDim.x + threadIdx.x; i < N; i += stride) {
        int v = (int)(i >> 9);               // K_N == 512
        Y[i] = (deg[v] - 1.0f) * PT[i] - Y[i];
    }
}

// ------------------------------- C = Q(512x8192) * Z(8192x512) via WMMA
__global__ void k_gemm(const float* __restrict__ Q, const float* __restrict__ Z,
                       float* __restrict__ C) {
    int lane = threadIdx.x & 31;
    int tile = (int)(((size_t)blockIdx.x * blockDim.x + threadIdx.x) >> 5);
    const int TPD = K_N / 16;                // 32 tiles per dim
    if (tile >= TPD * TPD) return;
    int mt = tile >> 5;
    int nt = tile & 31;
    int half = lane >> 4;                    // 0: K={0,1}, 1: K={2,3}
    int l16  = lane & 15;
    int ksel = half * 2;

    const float* qrow = Q + (size_t)(mt * 16 + l16) * V_N + ksel;
    const float* zcol = Z + (size_t)ksel * K_N + (nt * 16 + l16);

    v8f acc = {0.f, 0.f, 0.f, 0.f, 0.f, 0.f, 0.f, 0.f};
    for (int kk0 = 0; kk0 < V_N; kk0 += 64) {
        __builtin_prefetch(qrow + kk0 + 64, 0, 3);
        __builtin_prefetch(zcol + (size_t)(kk0 + 64) * K_N, 0, 3);
#pragma unroll
        for (int kk = kk0; kk < kk0 + 64; kk += 4) {
            v2f a, b;
            a.x = qrow[kk];
            a.y = qrow[kk + 1];
            b.x = zcol[(size_t)kk * K_N];
            b.y = zcol[(size_t)kk * K_N + K_N];
            acc = __builtin_amdgcn_wmma_f32_16x16x4_f32(
                false, a, false, b, (short)0, acc, false, false);
        }
    }
    // D layout: VGPR r -> M = r + 8*half, N = l16
    int mbase = mt * 16 + half * 8;
    int col   = nt * 16 + l16;
#pragma unroll
    for (int r = 0; r < 8; r++)
        C[(size_t)(mbase + r) * K_N + col] = acc[r];
}

// ------------- add EPS_L to diag, tr = sum(L_adj * S0^T), copy targets
__global__ void k_trace(float* __restrict__ C, const float* __restrict__ T,
                        float* __restrict__ Tc, float* __restrict__ trAcc) {
    __shared__ float red[256];
    const size_t N = (size_t)K_N * K_N;
    size_t stride = (size_t)gridDim.x * blockDim.x;
    float local = 0.0f;
    for (size_t i = (size_t)blockIdx.x * blockDim.x + threadIdx.x; i < N; i += stride) {
        int a = (int)(i >> 9);
        int b = (int)(i & (K_N - 1));
        float val = C[i];
        if (a == b) { val += EPSL; C[i] = val; }
        local += val * T[(size_t)b * K_N + a];
        Tc[i] = T[i];
    }
    red[threadIdx.x] = local;
    __syncthreads();
    for (int s = 128; s; s >>= 1) {
        if (threadIdx.x < s) red[threadIdx.x] += red[threadIdx.x + s];
        __syncthreads();
    }
    if (threadIdx.x == 0) atomicAdd(trAcc, red[0]);
}

// --------- single-block Gaussian elimination (SPD, no pivoting) -> logdet
__global__ void k_logdet(float* __restrict__ A, float* __restrict__ out) {
    __shared__ float s_inv;
    __shared__ float s_acc;
    const int n = K_N;
    if (threadIdx.x == 0) s_acc = 0.0f;
    for (int j = 0; j < n; j++) {
        __syncthreads();
        if (threadIdx.x == 0) {
            float piv = A[(size_t)j * n + j];
            s_acc += __logf(fabsf(piv));
            s_inv = 1.0f / piv;
        }
        __syncthreads();
        int m = n - 1 - j;
        float inv = s_inv;
        const float* prow = A + (size_t)j * n;
        for (int idx = threadIdx.x; idx < m * m; idx += blockDim.x) {
            int r = j + 1 + idx / m;
            int c = j + 1 + idx % m;
            A[(size_t)r * n + c] -= A[(size_t)r * n + j] * inv * prow[c];
        }
    }
    __syncthreads();
    if (threadIdx.x == 0) out[0] = s_acc;
}

// ------------------------------------------------------------- assemble
__global__ void k_final(const float* __restrict__ scal, float* __restrict__ out) {
    // scal[0]=tr, scal[1]=logdet(L_adj), scal[2]=logdet(targets)
    out[0] = -scal[1] - scal[2] - (float)K_N + scal[0];
}

extern "C" void kernel_launch(void* const* d_in, const int* in_sizes, int n_in,
                              void* d_out, int out_size, void* d_ws, size_t ws_size,
                              hipStream_t stream) {
    const float* inputs  = (const float*)d_in[0];   // V x 128
    const float* targets = (const float*)d_in[1];   // K x K
    const int*   eidx    = (const int*)d_in[2];     // 2 x E
    const float* pmat    = (const float*)d_in[3];   // K x V
    float* out = (float*)d_out;

    float* ws   = (float*)d_ws;
    float* deg  = ws;                                // V
    float* PT   = deg + V_N;                         // V*K
    float* Y    = PT + (size_t)V_N * K_N;            // V*K  (becomes Z)
    float* C    = Y + (size_t)V_N * K_N;             // K*K  (L_adj)
    float* Tc   = C + (size_t)K_N * K_N;             // K*K  (targets copy)
    float* scal = Tc + (size_t)K_N * K_N;            // 16 scalars

    // 1) init: Y=0, deg=1, scalars=0
    k_init<<<2048, 256, 0, stream>>>(Y, deg, scal);
    // 2) PT = pmat^T
    {
        dim3 g(V_N / 32, K_N / 32), b(32, 32);
        k_transpose<<<g, b, 0, stream>>>(pmat, PT);
    }
    // 3) edge weights + degree + scatter Y = S * PT   (8 waves / block)
    k_edges<<<E_N / 8, 256, 0, stream>>>(inputs, eidx, PT, Y, deg);
    // 4) Z = diag(deg-1)*PT - Y (in place)
    k_z<<<2048, 256, 0, stream>>>(Y, PT, deg);
    // 5) C = pmat * Z  (fp32 WMMA 16x16x4), 1024 tiles, 8 waves / block
    k_gemm<<<(K_N / 16) * (K_N / 16) / 8, 256, 0, stream>>>(pmat, Y, C);
    // 6) eps on diag, trace against targets^T, copy targets for elimination
    k_trace<<<256, 256, 0, stream>>>(C, targets, Tc, &scal[0]);
    // 7) logdets
    k_logdet<<<1, 1024, 0, stream>>>(C,  &scal[1]);
    k_logdet<<<1, 1024, 0, stream>>>(Tc, &scal[2]);
    // 8) final scalar
    k_final<<<1, 1, 0, stream>>>(scal, out);
}